// GCN_64931315581106
// MI455X (gfx1250) — compile-verified
//
#include <hip/hip_runtime.h>

typedef float v2f __attribute__((ext_vector_type(2)));
typedef float v8f __attribute__((ext_vector_type(8)));

#define FDIM 64

// ---------------------------------------------------------------- degree pass
__global__ __launch_bounds__(256) void k_fill1(float* __restrict__ p, int n) {
  int i = blockIdx.x * 256 + threadIdx.x;
  if (i < n) p[i] = 1.0f;  // self-loop contributes 1 to every degree
}

__global__ __launch_bounds__(256) void k_deg(const int* __restrict__ dst,
                                             float* __restrict__ deg, int e) {
  int i = blockIdx.x * 256 + threadIdx.x;
  if (i < e) {
    __hip_atomic_fetch_add(&deg[dst[i]], 1.0f,
                           __ATOMIC_RELAXED, __HIP_MEMORY_SCOPE_AGENT);
  }
}

__global__ __launch_bounds__(256) void k_rsqrt(float* __restrict__ p, int n) {
  int i = blockIdx.x * 256 + threadIdx.x;
  if (i < n) p[i] = rsqrtf(p[i]);  // deg >= 1 always (self-loop)
}

// ------------------------------------------------------------- fused WMMA GEMM
// Computes h = X @ W (fp32, V_WMMA_F32_16X16X4_F32, one wave per 16x16 tile),
// then in the register epilogue emits BOTH:
//   hs[row]  = h[row]  * dis[row]          (pre-scaled messages for scatter)
//   agg[row] = hs[row] * dis[row]          (self-loop term, = h/deg)
// A 16x4 f32: lanes 0-15 hold K={0,1} in VGPR{0,1}; lanes 16-31 hold K={2,3}.
// B 4x16 f32: VGPR0 = rows {K0 | K2}, VGPR1 = rows {K1 | K3}, N = lane&15.
// D 16x16 f32: VGPR r -> M = r + (lane>=16 ? 8 : 0), N = lane&15.
__global__ __launch_bounds__(256) void k_gemm_fused(const float* __restrict__ X,
                                                    const float* __restrict__ W,
                                                    const float* __restrict__ dis,
                                                    float* __restrict__ HS,
                                                    float* __restrict__ AGG,
                                                    int nrows) {
  int wave = (blockIdx.x * 256 + (int)threadIdx.x) >> 5;
  int lane = (int)threadIdx.x & 31;
  int tiles_m = (nrows + 15) >> 4;
  int tm = wave >> 2;       // 4 N-tiles (64/16)
  int tn = wave & 3;
  if (tm >= tiles_m) return;  // wave-uniform exit; EXEC stays all-1s for WMMA

  int hw = lane >> 4;       // half-wave select
  int l  = lane & 15;
  int rowA = tm * 16 + l;
  int rA   = rowA < nrows ? rowA : nrows - 1;  // clamp loads, keep EXEC full
  int colN = tn * 16 + l;

  v8f acc = {};
#pragma unroll
  for (int k0 = 0; k0 < FDIM; k0 += 4) {
    int ka = k0 + hw * 2;
    v2f a, b;
    a.x = X[(size_t)rA * FDIM + ka];
    a.y = X[(size_t)rA * FDIM + ka + 1];
    b.x = W[(size_t)(ka + 0) * FDIM + colN];
    b.y = W[(size_t)(ka + 1) * FDIM + colN];
    acc = __builtin_amdgcn_wmma_f32_16x16x4_f32(false, a, false, b,
                                                (short)0, acc, false, false);
  }

  int rbase = tm * 16 + hw * 8;
#pragma unroll
  for (int r = 0; r < 8; ++r) {
    int row = rbase + r;
    if (row < nrows) {
      float dr = dis[row];                 // uniform across the 16-lane half
      float hs = acc[r] * dr;
      HS [(size_t)row * FDIM + colN] = hs;
      AGG[(size_t)row * FDIM + colN] = hs * dr;   // self-loop: h / deg
    }
  }
}

// ---------------------------------------------------------------- edge scatter
// 16 lanes per edge, float4 per lane (b128 gathers); msg = hs[src] * dis[dst].
__global__ __launch_bounds__(256) void k_scatter(const float* __restrict__ HS,
                                                 const int* __restrict__ src,
                                                 const int* __restrict__ dst,
                                                 const float* __restrict__ dis,
                                                 float* __restrict__ agg, int nedges) {
  long long gid = (long long)blockIdx.x * 256 + threadIdx.x;
  int e    = (int)(gid >> 4);
  int lane = (int)(gid & 15);
  if (e >= nedges) return;
  int s = src[e];
  int d = dst[e];
  float nd = dis[d];
  float4 h = ((const float4*)(HS + (size_t)s * FDIM))[lane];
  float* ap = agg + (size_t)d * FDIM + lane * 4;
  __hip_atomic_fetch_add(ap + 0, h.x * nd, __ATOMIC_RELAXED, __HIP_MEMORY_SCOPE_AGENT);
  __hip_atomic_fetch_add(ap + 1, h.y * nd, __ATOMIC_RELAXED, __HIP_MEMORY_SCOPE_AGENT);
  __hip_atomic_fetch_add(ap + 2, h.z * nd, __ATOMIC_RELAXED, __HIP_MEMORY_SCOPE_AGENT);
  __hip_atomic_fetch_add(ap + 3, h.w * nd, __ATOMIC_RELAXED, __HIP_MEMORY_SCOPE_AGENT);
}

// ---------------------------------------------------------------- bias + relu
__global__ __launch_bounds__(256) void k_epilogue(const float* __restrict__ agg,
                                                  const float* __restrict__ bias,
                                                  float* __restrict__ out,
                                                  int n, int do_relu) {
  int i = blockIdx.x * 256 + threadIdx.x;      // over n*16 float4 chunks
  if (i >= n * (FDIM / 4)) return;
  int f = (i & 15) * 4;
  float4 v = ((const float4*)agg)[i];
  float4 b = *(const float4*)(bias + f);
  v.x += b.x; v.y += b.y; v.z += b.z; v.w += b.w;
  if (do_relu) {
    v.x = fmaxf(v.x, 0.0f); v.y = fmaxf(v.y, 0.0f);
    v.z = fmaxf(v.z, 0.0f); v.w = fmaxf(v.w, 0.0f);
  }
  ((float4*)out)[i] = v;
}

// layer: gemm_fused(in -> hs, agg) ; scatter(hs -> agg) ; epilogue(agg -> out)
// hs and agg are distinct buffers; `in` may alias `out` (hs is read-only after
// the gemm, epilogue only reads agg).
static void run_layer(const float* in, const float* W, const float* b,
                      float* hsbuf, float* aggbuf, const float* dis,
                      float* out, int N, int E,
                      const int* src, const int* dst, int relu,
                      hipStream_t stream) {
  int tiles = ((N + 15) / 16) * 4;
  int gemm_blocks = (tiles * 32 + 255) / 256;
  k_gemm_fused<<<gemm_blocks, 256, 0, stream>>>(in, W, dis, hsbuf, aggbuf, N);

  long long sthreads = (long long)E * 16;
  int sblocks = (int)((sthreads + 255) / 256);
  k_scatter<<<sblocks, 256, 0, stream>>>(hsbuf, src, dst, dis, aggbuf, E);

  int vec_blocks = (N * (FDIM / 4) + 255) / 256;
  k_epilogue<<<vec_blocks, 256, 0, stream>>>(aggbuf, b, out, N, relu);
}

extern "C" void kernel_launch(void* const* d_in, const int* in_sizes, int n_in,
                              void* d_out, int out_size, void* d_ws, size_t ws_size,
                              hipStream_t stream) {
  const float* x  = (const float*)d_in[0];
  const int*   ei = (const int*)d_in[1];
  const float* W1 = (const float*)d_in[2];
  const float* b1 = (const float*)d_in[3];
  const float* W2 = (const float*)d_in[4];
  const float* b2 = (const float*)d_in[5];
  const float* W3 = (const float*)d_in[6];
  const float* b3 = (const float*)d_in[7];

  int N = in_sizes[0] / FDIM;   // 100000
  int E = in_sizes[1] / 2;      // 1280000
  const int* src = ei;          // edge_index row 0
  const int* dst = ei + E;      // edge_index row 1

  // workspace: dis[N] | hs[N*64] | agg[N*64] | hio[N*64]  (~77 MB)
  float* dis = (float*)d_ws;
  float* hs  = dis + N;                        // 400000 B offset, 16B aligned
  float* agg = hs  + (size_t)N * FDIM;
  float* hio = agg + (size_t)N * FDIM;

  int nb = (N + 255) / 256;
  int eb = (E + 255) / 256;
  k_fill1<<<nb, 256, 0, stream>>>(dis, N);
  k_deg<<<eb, 256, 0, stream>>>(dst, dis, E);
  k_rsqrt<<<nb, 256, 0, stream>>>(dis, N);

  run_layer(x,   W1, b1, hs, agg, dis, hio,           N, E, src, dst, 1, stream);
  run_layer(hio, W2, b2, hs, agg, dis, hio,           N, E, src, dst, 1, stream);
  run_layer(hio, W3, b3, hs, agg, dis, (float*)d_out, N, E, src, dst, 0, stream);
}